// MSE_2d_Loss_25658134626813
// MI455X (gfx1250) — compile-verified
//
#include <hip/hip_runtime.h>
#include <hip/hip_bf16.h>

// ---------------------------------------------------------------------------
// OHEM MSE loss, MI455X (gfx1250, wave32).
//
// Pass 1 (bandwidth-bound, 64 MB streamed once ~= 2.8us at 23.3 TB/s):
//   grid = 32 rows x 32 chunks, 256 thr/block, float4 (b128) loads.
//   Per block: LDS histogram (4096 bins keyed on float bit-pattern >> 19,
//   monotone for loss >= 0; in-bin spread 2^-4) with int counts + u64
//   fixed-point (2^20) sums -> deterministic integer atomics only.
//   Per-wave partial sums (sumLoss, sumPosLoss, npos) reduced with the
//   matrix engine: V_WMMA_F32_16X16X4_F32(A=[v,0], B=ones) gives
//   D[m][n] = v(m)+v(m+16); sum of the 8 D VGPRs + shfl_xor(16) = wave sum.
//   Partials stored to fixed scratch slots (no float atomics -> deterministic).
//
// Pass 2 (32 blocks): deterministic tree-reduce of 256 partials/row, then
//   serial suffix scan of the 4096-bin histogram to find the k-th largest
//   negative loss; top-k sum = exact suffix sum + remainder * in-bin mean.
//
// Pass 3 (1 wave): fixed-order WMMA wave reduction of 32 per-sample values.
// ---------------------------------------------------------------------------

typedef float v2f __attribute__((ext_vector_type(2)));
typedef float v8f __attribute__((ext_vector_type(8)));

#define B_      32
#define N_      262144          // 512*512
#define NBINS   4096            // __float_as_uint(loss) >> 19  (max 4080)
#define BPR     32              // pass-1 blocks per row
#define TPB     256
#define WPB     (TPB / 32)      // 8 waves per block
#define PARTS   (BPR * WPB)     // 256 wave-partials per row
#define QSCALE  1048576.0f      // 2^20 fixed point for deterministic u64 sums
#define QINV    (1.0 / 1048576.0)

// Full wave32 sum using the matrix engine + one cross-half shuffle.
// EXEC must be all ones at the call site (callers are fully convergent).
__device__ __forceinline__ float wave_reduce_wmma(float v) {
    v2f a; a[0] = v;    a[1] = 0.0f;    // A[m][0]=v(m), A[m][2]=v(m+16)
    v2f b; b[0] = 1.0f; b[1] = 1.0f;    // B = ones
    v8f c = {};
    // (neg_a, A, neg_b, B, c_mod, C, reuse_a, reuse_b)
    v8f d = __builtin_amdgcn_wmma_f32_16x16x4_f32(false, a, false, b,
                                                  (short)0, c, false, false);
    // lanes 0-15 hold rows 0..7 -> sum = v(0..7)+v(16..23)
    // lanes 16-31 hold rows 8..15 -> sum = v(8..15)+v(24..31)
    float s = d[0] + d[1] + d[2] + d[3] + d[4] + d[5] + d[6] + d[7];
    s += __shfl_xor(s, 16, 32);
    return s;
}

__global__ __launch_bounds__(TPB)
void ohem_pass1(const float* __restrict__ x, const float* __restrict__ y,
                float* __restrict__ pLoss, float* __restrict__ pPos,
                float* __restrict__ pNpos,
                unsigned long long* __restrict__ gSum, int* __restrict__ gCnt) {
    __shared__ int                hc[NBINS];   // 16 KB
    __shared__ unsigned long long hs[NBINS];   // 32 KB

    const int tid   = threadIdx.x;
    const int row   = blockIdx.x / BPR;
    const int chunk = blockIdx.x % BPR;

    for (int i = tid; i < NBINS; i += TPB) { hc[i] = 0; hs[i] = 0ull; }
    __syncthreads();

    const float4* x4 = (const float4*)x;
    const float4* y4 = (const float4*)y;
    const int vec4PerBlock = (N_ / BPR) / 4;                 // 2048
    const int base4 = row * (N_ / 4) + chunk * vec4PerBlock;

    float sumLoss = 0.0f, sumPos = 0.0f, nposF = 0.0f;

    #pragma unroll
    for (int it = 0; it < vec4PerBlock / TPB; ++it) {        // 8 iterations
        const int off = base4 + it * TPB + tid;
        float4 xv = x4[off];                                  // global_load_b128
        float4 yv = y4[off];
        #pragma unroll
        for (int cI = 0; cI < 4; ++cI) {
            float xe = (&xv.x)[cI];
            float ye = (&yv.x)[cI];
            float df = xe - ye;
            float l  = df * df;
            sumLoss += l;
            if (ye > 0.0f) {
                sumPos += l;
                nposF  += 1.0f;
            } else {
                unsigned bin = __float_as_uint(l) >> 19;      // monotone key
                atomicAdd(&hc[bin], 1);                       // ds_add
                atomicAdd(&hs[bin], (unsigned long long)(l * QSCALE)); // ds_add_u64
            }
        }
    }

    // Deterministic wave reductions on the matrix engine (all lanes active).
    float rl = wave_reduce_wmma(sumLoss);
    float rp = wave_reduce_wmma(sumPos);
    float rn = wave_reduce_wmma(nposF);
    if ((tid & 31) == 0) {
        const int slot = chunk * WPB + (tid >> 5);            // fixed slot
        pLoss[row * PARTS + slot] = rl;
        pPos [row * PARTS + slot] = rp;
        pNpos[row * PARTS + slot] = rn;
    }

    __syncthreads();  // LDS histogram complete before flush
    for (int i = tid; i < NBINS; i += TPB) {
        const int c = hc[i];
        if (c) {
            atomicAdd(&gCnt[row * NBINS + i], c);             // integer: order-invariant
            atomicAdd(&gSum[row * NBINS + i], hs[i]);         // u64: order-invariant
        }
    }
}

__global__ __launch_bounds__(TPB)
void ohem_pass2(const float* __restrict__ pLoss, const float* __restrict__ pPos,
                const float* __restrict__ pNpos,
                const unsigned long long* __restrict__ gSum,
                const int* __restrict__ gCnt, float* __restrict__ perSample) {
    __shared__ float rA[TPB], rB[TPB], rC[TPB];
    __shared__ long long          chCnt[TPB];
    __shared__ unsigned long long chSum[TPB];

    const int tid = threadIdx.x;
    const int row = blockIdx.x;
    const int per = NBINS / TPB;                              // 16 bins/thread

    rA[tid] = pLoss[row * PARTS + tid];
    rB[tid] = pPos [row * PARTS + tid];
    rC[tid] = pNpos[row * PARTS + tid];

    long long cc = 0; unsigned long long cs = 0;
    for (int j = 0; j < per; ++j) {
        const int bin = tid * per + j;
        cc += gCnt[row * NBINS + bin];
        cs += gSum[row * NBINS + bin];
    }
    chCnt[tid] = cc; chSum[tid] = cs;
    __syncthreads();

    // Deterministic block tree reduction.
    for (int s = TPB / 2; s > 0; s >>= 1) {
        if (tid < s) {
            rA[tid] += rA[tid + s];
            rB[tid] += rB[tid + s];
            rC[tid] += rC[tid + s];
        }
        __syncthreads();
    }

    if (tid == 0) {
        const float sumLoss = rA[0], sumPos = rB[0], nposF = rC[0];
        const int npos = (int)(nposF + 0.5f);
        const int k    = (int)(nposF * 3.0f);                 // trunc, matches ref
        const bool cond = ((long long)k + npos >= N_) || (k <= 10);
        float perV;
        if (cond) {
            perV = sumLoss / (float)N_;
        } else {
            const float posMean = sumPos / (float)(npos > 1 ? npos : 1);
            const long long need = k;
            long long cum = 0; unsigned long long above = 0;
            int chunkI = TPB - 1;
            for (; chunkI >= 0; --chunkI) {                   // suffix scan (top down)
                if (cum + chCnt[chunkI] >= need) break;
                cum += chCnt[chunkI]; above += chSum[chunkI];
            }
            double topkQ;
            if (chunkI < 0) {
                topkQ = (double)above;                        // k >= #neg (guarded by cond)
            } else {
                const int binLo = chunkI * per;
                int bin; long long cT = 0;
                for (bin = binLo + per - 1; bin >= binLo; --bin) {
                    const long long c = gCnt[row * NBINS + bin];
                    if (cum + c >= need) { cT = c; break; }
                    cum += c; above += gSum[row * NBINS + bin];
                }
                if (bin < binLo) { bin = binLo; cT = gCnt[row * NBINS + bin]; }
                long long r = need - cum;
                if (r < 0) r = 0; if (r > cT) r = cT;
                const double avg =
                    (cT > 0) ? (double)gSum[row * NBINS + bin] / (double)cT : 0.0;
                topkQ = (double)above + avg * (double)r;      // tie-bin: in-bin mean
            }
            const float negMean = (float)(topkQ * QINV) / (float)(k > 1 ? k : 1);
            perV = posMean + negMean;
        }
        perSample[row] = perV;
    }
}

__global__ __launch_bounds__(32)
void ohem_pass3(const float* __restrict__ perSample, float* __restrict__ out) {
    const float v = perSample[threadIdx.x];      // 32 threads = 1 full wave
    const float s = wave_reduce_wmma(v);         // fixed-order -> deterministic
    if (threadIdx.x == 0) out[0] = s * (1.0f / 32.0f);
}

extern "C" void kernel_launch(void* const* d_in, const int* in_sizes, int n_in,
                              void* d_out, int out_size, void* d_ws, size_t ws_size,
                              hipStream_t stream) {
    const float* x = (const float*)d_in[0];
    const float* y = (const float*)d_in[1];

    char* ws = (char*)d_ws;
    size_t o = 0;
    float* pLoss = (float*)(ws + o); o += (size_t)B_ * PARTS * sizeof(float);
    float* pPos  = (float*)(ws + o); o += (size_t)B_ * PARTS * sizeof(float);
    float* pNpos = (float*)(ws + o); o += (size_t)B_ * PARTS * sizeof(float);
    o = (o + 15) & ~(size_t)15;
    unsigned long long* gSum = (unsigned long long*)(ws + o);
    o += (size_t)B_ * NBINS * sizeof(unsigned long long);
    int* gCnt = (int*)(ws + o); o += (size_t)B_ * NBINS * sizeof(int);
    float* perSample = (float*)(ws + o); o += (size_t)B_ * sizeof(float);
    // total ~1.7 MB of scratch

    hipMemsetAsync(d_ws, 0, o, stream);          // zero hist + stats (capture-safe)

    ohem_pass1<<<B_ * BPR, TPB, 0, stream>>>(x, y, pLoss, pPos, pNpos, gSum, gCnt);
    ohem_pass2<<<B_, TPB, 0, stream>>>(pLoss, pPos, pNpos, gSum, gCnt, perSample);
    ohem_pass3<<<1, 32, 0, stream>>>(perSample, (float*)d_out);
}